// GridSampling3D_67997922230611
// MI455X (gfx1250) — compile-verified
//
#include <hip/hip_runtime.h>
#include <hip/hip_bf16.h>
#include <stdint.h>

// ---------------------------------------------------------------------------
// GridSampling3D for MI455X (gfx1250, wave32).
// Deterministic pipeline: minmax -> params -> keys -> 3x stable radix sort
// (11+10+10 bits) -> head-flag scan (inverse indices) -> segment mean.
// Uses gfx1250 async global->LDS staging + global_prefetch.
// ---------------------------------------------------------------------------

#define THREADS   256
#define IPT       16
#define TILE      (THREADS * IPT)   // 4096 elements per block

// -------- gfx1250 async global->LDS staging (probe-confirmed builtins) ------
#ifndef HAVE_GFX1250_ASYNC
#if defined(__has_builtin)
#if __has_builtin(__builtin_amdgcn_global_load_async_to_lds_b32) && \
    __has_builtin(__builtin_amdgcn_s_wait_asynccnt)
#define HAVE_GFX1250_ASYNC 1
#endif
#endif
#endif
#ifndef HAVE_GFX1250_ASYNC
#define HAVE_GFX1250_ASYNC 0
#endif

// -------- monotone float<->uint order-preserving map ------------------------
__device__ __forceinline__ uint32_t ford(float f) {
  uint32_t u = __float_as_uint(f);
  return (u & 0x80000000u) ? ~u : (u | 0x80000000u);
}
__device__ __forceinline__ float funord(uint32_t u) {
  u = (u & 0x80000000u) ? (u & 0x7FFFFFFFu) : ~u;
  return __uint_as_float(u);
}

// hdr layout (uint32 slots): [0..3] min(ord), [4..7] max(ord),
// [8..11] start (float), [12..15] num_vox (int), [20..23] strides (int),
// [24] num_unique
__global__ __launch_bounds__(32) void gs_init(uint32_t* hdr) {
  int t = threadIdx.x;
  if (t < 4) { hdr[t] = 0xFFFFFFFFu; hdr[4 + t] = 0u; }
  if (t == 0) hdr[24] = 0u;
}

__global__ __launch_bounds__(THREADS) void gs_minmax(const float4* __restrict__ pts,
                                                     int N, uint32_t* hdr) {
  __shared__ uint32_t lmin[4], lmax[4];
  int t = threadIdx.x;
  if (t < 4) { lmin[t] = 0xFFFFFFFFu; lmax[t] = 0u; }
  __syncthreads();
  uint32_t mn[4] = {~0u, ~0u, ~0u, ~0u};
  uint32_t mx[4] = {0u, 0u, 0u, 0u};
  int base = blockIdx.x * TILE;
#pragma unroll
  for (int j = 0; j < IPT; ++j) {
    int i = base + j * THREADS + t;
    if (i < N) {
      float4 p = pts[i];
      uint32_t v0 = ford(p.x), v1 = ford(p.y), v2 = ford(p.z), v3 = ford(p.w);
      mn[0] = min(mn[0], v0); mx[0] = max(mx[0], v0);
      mn[1] = min(mn[1], v1); mx[1] = max(mx[1], v1);
      mn[2] = min(mn[2], v2); mx[2] = max(mx[2], v2);
      mn[3] = min(mn[3], v3); mx[3] = max(mx[3], v3);
    }
  }
#pragma unroll
  for (int c = 0; c < 4; ++c) {
    atomicMin(&lmin[c], mn[c]);
    atomicMax(&lmax[c], mx[c]);
  }
  __syncthreads();
  if (t < 4) {
    atomicMin(&hdr[t], lmin[t]);
    atomicMax(&hdr[4 + t], lmax[t]);
  }
}

__global__ void gs_params(uint32_t* hdr) {
  if (threadIdx.x != 0 || blockIdx.x != 0) return;
  const float g[4] = {1.0f, 0.1f, 0.1f, 0.1f};
  float st[4], en[4];
  int nv[4], sd[4];
#pragma unroll
  for (int c = 0; c < 4; ++c) { st[c] = funord(hdr[c]); en[c] = funord(hdr[4 + c]); }
  st[0] -= 0.5f;
  en[0] += 0.5f;
#pragma unroll
  for (int c = 0; c < 4; ++c) nv[c] = (int)floorf((en[c] - st[c]) / g[c]) + 1;
  sd[0] = 1;
  sd[1] = nv[0];
  sd[2] = nv[0] * nv[1];
  sd[3] = sd[2] * nv[2];
  float* hf = (float*)hdr;
  int* hi = (int*)hdr;
#pragma unroll
  for (int c = 0; c < 4; ++c) { hf[8 + c] = st[c]; hi[12 + c] = nv[c]; hi[20 + c] = sd[c]; }
}

__global__ __launch_bounds__(THREADS) void gs_makekeys(const float4* __restrict__ pts,
                                                       int N, const uint32_t* hdr,
                                                       uint32_t* __restrict__ keys,
                                                       uint32_t* __restrict__ idx) {
  const float* hf = (const float*)hdr;
  const int* hi = (const int*)hdr;
  float s0 = hf[8], s1 = hf[9], s2 = hf[10], s3 = hf[11];
  int d0 = hi[20], d1 = hi[21], d2 = hi[22], d3 = hi[23];
  int t = threadIdx.x;
  int base = blockIdx.x * TILE;
#pragma unroll
  for (int j = 0; j < IPT; ++j) {
    int i = base + j * THREADS + t;
    if (i + 2 * THREADS < N) __builtin_prefetch(pts + i + 2 * THREADS, 0, 1);  // global_prefetch_b8
    if (i < N) {
      float4 p = pts[i];
      int c0 = (int)floorf((p.x - s0) / 1.0f);
      int c1 = (int)floorf((p.y - s1) / 0.1f);
      int c2 = (int)floorf((p.z - s2) / 0.1f);
      int c3 = (int)floorf((p.w - s3) / 0.1f);
      int cluster = c0 * d0 + c1 * d1 + c2 * d2 + c3 * d3;  // int32, same as reference
      keys[i] = (uint32_t)cluster;
      idx[i] = (uint32_t)i;
    }
  }
}

// ------------------------- radix sort: histogram ----------------------------
template <int BITS>
__global__ __launch_bounds__(THREADS) void gs_hist(const uint32_t* __restrict__ ks,
                                                   int N, int nb, int shift,
                                                   uint32_t* __restrict__ hist) {
  constexpr int ND = 1 << BITS;
  __shared__ uint32_t h[ND];
  int t = threadIdx.x;
  for (int k = t; k < ND; k += THREADS) h[k] = 0u;
  __syncthreads();
  int base = blockIdx.x * TILE;
#pragma unroll
  for (int j = 0; j < IPT; ++j) {
    int g = base + j * THREADS + t;
    if (g + TILE < N) __builtin_prefetch(ks + g + TILE, 0, 1);
    if (g < N) atomicAdd(&h[(ks[g] >> shift) & (ND - 1)], 1u);
  }
  __syncthreads();
  for (int k = t; k < ND; k += THREADS)
    hist[(size_t)k * nb + blockIdx.x] = h[k];
}

// single-block exclusive scan (in place), optional grand total out
__global__ __launch_bounds__(1024) void gs_scan(uint32_t* data, int M, uint32_t* totalOut) {
  __shared__ uint32_t ts[1024];
  int t = threadIdx.x;
  int chunk = (M + 1023) / 1024;
  int beg = t * chunk;
  int end = min(beg + chunk, M);
  uint32_t s = 0;
  for (int i = beg; i < end; ++i) s += data[i];
  ts[t] = s;
  __syncthreads();
  for (int off = 1; off < 1024; off <<= 1) {
    uint32_t v = (t >= off) ? ts[t - off] : 0u;
    __syncthreads();
    ts[t] += v;
    __syncthreads();
  }
  uint32_t run = ts[t] - s;  // exclusive base for this thread
  for (int i = beg; i < end; ++i) {
    uint32_t v = data[i];
    data[i] = run;
    run += v;
  }
  if (totalOut && t == 1023) *totalOut = ts[1023];
}

// ---------------- radix sort: stable ballot-multisplit scatter --------------
template <int BITS>
__global__ __launch_bounds__(THREADS) void gs_scatter(const uint32_t* __restrict__ ks,
                                                      const uint32_t* __restrict__ vs,
                                                      int N, int nb, int shift,
                                                      const uint32_t* __restrict__ histScan,
                                                      uint32_t* __restrict__ kd,
                                                      uint32_t* __restrict__ vd) {
  constexpr int ND = 1 << BITS;
  __shared__ uint32_t cnt[ND];       // running global offset per digit
  __shared__ uint32_t wh[8 * ND];    // per-wave per-digit counts for one round
  __shared__ uint32_t lkey[TILE];
  __shared__ uint32_t lval[TILE];
  int t = threadIdx.x;
  int b = blockIdx.x;
  for (int k = t; k < ND; k += THREADS) cnt[k] = histScan[(size_t)k * nb + b];
  for (int k = t; k < 8 * ND; k += THREADS) wh[k] = 0u;
  int base = b * TILE;

  // ---- stage tile into LDS (gfx1250 async global->LDS DMA) ----
#if HAVE_GFX1250_ASYNC
  typedef __attribute__((address_space(1))) int GI;   // builtin arg0: as(1) int*
  typedef __attribute__((address_space(3))) int LI;   // builtin arg1: as(3) int*
#pragma unroll
  for (int j = 0; j < IPT; ++j) {
    int e = j * THREADS + t;
    int g = base + e;
    if (g < N) {
      __builtin_amdgcn_global_load_async_to_lds_b32((GI*)(ks + g), (LI*)&lkey[e], 0, 0);
      __builtin_amdgcn_global_load_async_to_lds_b32((GI*)(vs + g), (LI*)&lval[e], 0, 0);
    }
  }
  __builtin_amdgcn_s_wait_asynccnt(0);
#else
#pragma unroll
  for (int j = 0; j < IPT; ++j) {
    int e = j * THREADS + t;
    int g = base + e;
    if (g < N) {
      lkey[e] = ks[g];
      lval[e] = vs[g];
    }
  }
#endif
  __syncthreads();

  int lane = t & 31;
  int wv = t >> 5;
  uint32_t laneLT = (1u << lane) - 1u;

  for (int j = 0; j < IPT; ++j) {   // rounds in memory order -> stable
    int e = j * THREADS + t;
    int g = base + e;
    bool act = g < N;
    uint32_t key = act ? lkey[e] : 0u;
    uint32_t d = (key >> shift) & (uint32_t)(ND - 1);
    uint32_t peers = (uint32_t)__ballot(act);
#pragma unroll
    for (int bit = 0; bit < BITS; ++bit) {
      uint32_t bal = (uint32_t)__ballot(act && ((d >> bit) & 1u));
      peers &= ((d >> bit) & 1u) ? bal : ~bal;
    }
    uint32_t rIn = (uint32_t)__builtin_popcount(peers & laneLT);
    uint32_t gsz = (uint32_t)__builtin_popcount(peers);
    bool leader = act && (rIn == 0u);
    if (leader) wh[wv * ND + d] = gsz;  // group leader publishes count
    __syncthreads();
    if (act) {
      uint32_t off = cnt[d] + rIn;
      for (int w2 = 0; w2 < wv; ++w2) off += wh[w2 * ND + d];
      kd[off] = key;
      vd[off] = lval[e];
    }
    __syncthreads();
    if (leader) {
      atomicAdd(&cnt[d], gsz);  // integer add: deterministic regardless of order
      wh[wv * ND + d] = 0u;     // clear own entry for next round
    }
    __syncthreads();
  }
}

// ------------------- head flags + two-level rank scan -----------------------
__global__ __launch_bounds__(THREADS) void gs_rank_local(const uint32_t* __restrict__ keys,
                                                         int N, uint32_t* __restrict__ ranks,
                                                         uint32_t* __restrict__ blockSums) {
  __shared__ uint32_t ts[THREADS];
  int t = threadIdx.x;
  int base = blockIdx.x * TILE + t * IPT;  // blocked: serial scan order == memory order
  uint32_t s = 0;
#pragma unroll
  for (int j = 0; j < IPT; ++j) {
    int i = base + j;
    if (i < N) {
      uint32_t h = (i == 0 || keys[i] != keys[i - 1]) ? 1u : 0u;
      ranks[i] = h;
      s += h;
    }
  }
  ts[t] = s;
  __syncthreads();
  for (int off = 1; off < THREADS; off <<= 1) {
    uint32_t v = (t >= off) ? ts[t - off] : 0u;
    __syncthreads();
    ts[t] += v;
    __syncthreads();
  }
  uint32_t run = ts[t] - s;  // exclusive within block
  if (t == THREADS - 1) blockSums[blockIdx.x] = ts[THREADS - 1];
#pragma unroll
  for (int j = 0; j < IPT; ++j) {
    int i = base + j;
    if (i < N) {
      uint32_t h = ranks[i];
      ranks[i] = run;
      run += h;
    }
  }
}

// inv[idx[i]] = rank; ranks[i] <- global rank. inv stored as float (output
// buffer dtype is float32; ranks <= N fit exactly).
__global__ __launch_bounds__(THREADS) void gs_rank_final(const uint32_t* __restrict__ keys,
                                                         const uint32_t* __restrict__ idx,
                                                         int N, uint32_t* __restrict__ ranks,
                                                         const uint32_t* __restrict__ blockOff,
                                                         float* __restrict__ inv_out) {
  int t = threadIdx.x;
  uint32_t boff = blockOff[blockIdx.x];
#pragma unroll
  for (int j = 0; j < IPT; ++j) {
    int i = blockIdx.x * TILE + j * THREADS + t;
    if (i < N) {
      uint32_t h = (i == 0 || keys[i] != keys[i - 1]) ? 1u : 0u;
      uint32_t r = boff + ranks[i] + h - 1u;
      inv_out[idx[i]] = (float)r;
      ranks[i] = r;
    }
  }
}

// --------------------- contiguous segment mean ------------------------------
__global__ __launch_bounds__(THREADS) void gs_segmean(const uint32_t* __restrict__ keys,
                                                      const uint32_t* __restrict__ idx,
                                                      const float4* __restrict__ pts,
                                                      int N, const uint32_t* __restrict__ ranks,
                                                      float4* __restrict__ out) {
  int i = blockIdx.x * blockDim.x + threadIdx.x;
  if (i >= N) return;
  uint32_t k0 = keys[i];
  if (i != 0 && keys[i - 1] == k0) return;  // only segment heads
  float sx = 0.f, sy = 0.f, sz = 0.f, sw = 0.f;
  int c = 0;
  int j = i;
  while (j < N && keys[j] == k0) {  // ascending original index (stable sort)
    float4 p = pts[idx[j]];
    sx += p.x; sy += p.y; sz += p.z; sw += p.w;
    ++c;
    ++j;
  }
  float fc = (float)c;  // >= 1
  float4 o;
  o.x = sx / fc; o.y = sy / fc; o.z = sz / fc; o.w = sw / fc;
  out[ranks[i]] = o;
}

// ---------------------------------------------------------------------------
extern "C" void kernel_launch(void* const* d_in, const int* in_sizes, int n_in,
                              void* d_out, int out_size, void* d_ws, size_t ws_size,
                              hipStream_t stream) {
  const float4* pts = (const float4*)d_in[0];
  int N = in_sizes[0] / 4;
  int nb = (N + TILE - 1) / TILE;

  float* out = (float*)d_out;                 // [N,4] sampled grids
  float* inv_out = out + (size_t)4 * N;       // [N] inverse indices (as f32)

  uint8_t* ws = (uint8_t*)d_ws;
  uint32_t* hdr = (uint32_t*)ws;              // 256 B header
  uint32_t* keysA = (uint32_t*)(ws + 256);
  uint32_t* idxA = keysA + N;
  uint32_t* keysB = idxA + N;
  uint32_t* idxB = keysB + N;
  uint32_t* hist = idxB + N;                  // up to 2048*nb u32

  (void)hipMemsetAsync(d_out, 0, (size_t)out_size * sizeof(float), stream);

  gs_init<<<1, 32, 0, stream>>>(hdr);
  gs_minmax<<<nb, THREADS, 0, stream>>>(pts, N, hdr);
  gs_params<<<1, 1, 0, stream>>>(hdr);
  gs_makekeys<<<nb, THREADS, 0, stream>>>(pts, N, hdr, keysA, idxA);

  // 31 bits of key in 3 stable passes: 11 + 10 + 10
  uint32_t *ks = keysA, *vs = idxA, *kd = keysB, *vd = idxB;

  // pass 0: bits [10:0], 2048 buckets
  gs_hist<11><<<nb, THREADS, 0, stream>>>(ks, N, nb, 0, hist);
  gs_scan<<<1, 1024, 0, stream>>>(hist, 2048 * nb, nullptr);
  gs_scatter<11><<<nb, THREADS, 0, stream>>>(ks, vs, N, nb, 0, hist, kd, vd);
  { uint32_t* tk = ks; ks = kd; kd = tk; uint32_t* tv = vs; vs = vd; vd = tv; }

  // pass 1: bits [20:11], 1024 buckets
  gs_hist<10><<<nb, THREADS, 0, stream>>>(ks, N, nb, 11, hist);
  gs_scan<<<1, 1024, 0, stream>>>(hist, 1024 * nb, nullptr);
  gs_scatter<10><<<nb, THREADS, 0, stream>>>(ks, vs, N, nb, 11, hist, kd, vd);
  { uint32_t* tk = ks; ks = kd; kd = tk; uint32_t* tv = vs; vs = vd; vd = tv; }

  // pass 2: bits [30:21], 1024 buckets
  gs_hist<10><<<nb, THREADS, 0, stream>>>(ks, N, nb, 21, hist);
  gs_scan<<<1, 1024, 0, stream>>>(hist, 1024 * nb, nullptr);
  gs_scatter<10><<<nb, THREADS, 0, stream>>>(ks, vs, N, nb, 21, hist, kd, vd);
  { uint32_t* tk = ks; ks = kd; kd = tk; uint32_t* tv = vs; vs = vd; vd = tv; }

  // sorted keys/idx now in (ks, vs); the other pair is free scratch
  uint32_t* ranks = kd;
  uint32_t* blockSums = vd;

  gs_rank_local<<<nb, THREADS, 0, stream>>>(ks, N, ranks, blockSums);
  gs_scan<<<1, 1024, 0, stream>>>(blockSums, nb, &hdr[24]);
  gs_rank_final<<<nb, THREADS, 0, stream>>>(ks, vs, N, ranks, blockSums, inv_out);
  gs_segmean<<<(N + THREADS - 1) / THREADS, THREADS, 0, stream>>>(ks, vs, pts, N, ranks,
                                                                  (float4*)out);
}